// GCN2_23055384445766
// MI455X (gfx1250) — compile-verified
//
#include <hip/hip_runtime.h>
#include <cmath>

#define N_NODES   100000
#define N_FEAT    500
#define HIDDEN    64
#define N_CLASSES 40
#define N_LAYERS  8
#define ALPHA     0.1f

typedef __attribute__((ext_vector_type(2))) float v2f;
typedef __attribute__((ext_vector_type(8))) float v8f;

// ---------------------------------------------------------------------------
// Kernel 1: h = x0 = relu(x @ lin0_w + b).  One block = 16 rows, 4 waves,
// each wave owns a 16x16 column tile of HIDDEN=64. A tile staged in LDS.
// WMMA: V_WMMA_F32_16X16X4_F32, K looped 500/4 = 125 steps.
// ---------------------------------------------------------------------------
__global__ void __launch_bounds__(128)
lin0_kernel(const float* __restrict__ x, const float* __restrict__ w,
            const float* __restrict__ bias, float* __restrict__ h,
            float* __restrict__ x0)
{
    __shared__ float As[16 * N_FEAT];   // 32 KB
    const int tid   = threadIdx.x;
    const int wave  = tid >> 5;
    const int lane  = tid & 31;
    const int row0  = blockIdx.x * 16;
    const int n0    = wave * 16;

    // cooperative, coalesced load of the 16 x 500 A tile
    for (int r = 0; r < 16; ++r)
        for (int c = tid; c < N_FEAT; c += 128)
            As[r * N_FEAT + c] = x[(size_t)(row0 + r) * N_FEAT + c];
    __syncthreads();

    const int m     = lane & 15;
    const int khalf = lane >> 4;       // 0: K={0,1}, 1: K={2,3}
    const int col   = lane & 15;

    v8f acc = {};
    for (int k0 = 0; k0 < N_FEAT; k0 += 4) {
        const int ka = k0 + khalf * 2;
        v2f a, b;
        a.x = As[m * N_FEAT + ka];
        a.y = As[m * N_FEAT + ka + 1];
        b.x = w[(size_t)ka       * HIDDEN + n0 + col];
        b.y = w[(size_t)(ka + 1) * HIDDEN + n0 + col];
        acc = __builtin_amdgcn_wmma_f32_16x16x4_f32(false, a, false, b,
                                                    (short)0, acc, false, false);
    }

    const float bv = bias[n0 + col];
#pragma unroll
    for (int r = 0; r < 8; ++r) {
        const int   mrow = r + khalf * 8;
        const float v    = fmaxf(acc[r] + bv, 0.0f);
        const size_t idx = (size_t)(row0 + mrow) * HIDDEN + n0 + col;
        h[idx]  = v;
        x0[idx] = v;
    }
}

// ---------------------------------------------------------------------------
// Kernel 2a: zero agg buffer (graph-safe, no memset)
// ---------------------------------------------------------------------------
__global__ void zero_kernel(float* __restrict__ p, int n)
{
    int stride = gridDim.x * blockDim.x;
    for (int i = blockIdx.x * blockDim.x + threadIdx.x; i < n; i += stride)
        p[i] = 0.0f;
}

// ---------------------------------------------------------------------------
// Kernel 2b: SpMM scatter: agg[dst] += w * h[src]. One edge per wave32:
// each lane handles 2 of 64 features (contiguous 256B gather + atomic scatter,
// both serviced from the 192MB L2 since h/agg are 25.6MB each).
// ---------------------------------------------------------------------------
__global__ void spmm_kernel(const int* __restrict__ esrc,
                            const int* __restrict__ edst,
                            const float* __restrict__ ew,
                            const float* __restrict__ h,
                            float* __restrict__ agg, int nE)
{
    const int lane   = threadIdx.x & 31;
    const int wid    = (blockIdx.x * blockDim.x + threadIdx.x) >> 5;
    const int nwaves = (gridDim.x * blockDim.x) >> 5;

    for (int e = wid; e < nE; e += nwaves) {
        const int   s  = esrc[e];
        const int   d  = edst[e];
        const float wt = ew[e];
        if (e + nwaves < nE) {
            __builtin_prefetch(&esrc[e + nwaves], 0, 0);   // global_prefetch_b8
        }
        const float2 hv = ((const float2*)(h + (size_t)s * HIDDEN))[lane];
        float* dp = agg + (size_t)d * HIDDEN + lane * 2;
        atomicAdd(dp,     wt * hv.x);
        atomicAdd(dp + 1, wt * hv.y);
    }
}

// ---------------------------------------------------------------------------
// Kernel 3: one GCNII layer combine:
//   z = 0.9*agg + 0.1*x0 ; h = relu((1-beta)*z + beta*(z @ W))
// z tile (16x64) staged in LDS (stride 68 to avoid bank conflicts on the
// A-fragment column reads), WMMA 16x16x4, K=64 -> 16 steps per tile.
// Writes back into h (reads only agg/x0), so no extra buffer needed.
// ---------------------------------------------------------------------------
#define ZLD 68
__global__ void __launch_bounds__(128)
layer_kernel(const float* __restrict__ agg, const float* __restrict__ x0,
             const float* __restrict__ w, float* __restrict__ h, float beta)
{
    __shared__ float Zs[16 * ZLD];
    const int tid  = threadIdx.x;
    const int wave = tid >> 5;
    const int lane = tid & 31;
    const int row0 = blockIdx.x * 16;
    const int n0   = wave * 16;

    for (int i = tid; i < 16 * HIDDEN; i += 128) {
        const int r = i >> 6, c = i & 63;
        const size_t g = (size_t)(row0 + r) * HIDDEN + c;
        Zs[r * ZLD + c] = (1.0f - ALPHA) * agg[g] + ALPHA * x0[g];
    }
    __syncthreads();

    const int m     = lane & 15;
    const int khalf = lane >> 4;
    const int col   = lane & 15;

    v8f acc = {};
#pragma unroll
    for (int k0 = 0; k0 < HIDDEN; k0 += 4) {
        const int ka = k0 + khalf * 2;
        v2f a, b;
        a.x = Zs[m * ZLD + ka];
        a.y = Zs[m * ZLD + ka + 1];
        b.x = w[(ka)     * HIDDEN + n0 + col];
        b.y = w[(ka + 1) * HIDDEN + n0 + col];
        acc = __builtin_amdgcn_wmma_f32_16x16x4_f32(false, a, false, b,
                                                    (short)0, acc, false, false);
    }

#pragma unroll
    for (int r = 0; r < 8; ++r) {
        const int   mrow = r + khalf * 8;
        const float zv   = Zs[mrow * ZLD + n0 + col];
        const float v    = (1.0f - beta) * zv + beta * acc[r];
        h[(size_t)(row0 + mrow) * HIDDEN + n0 + col] = fmaxf(v, 0.0f);
    }
}

// ---------------------------------------------------------------------------
// Kernel 4: out = h @ lin1_w + b   (100k x 64 x 40). 3 waves per block
// (col tiles 0..47, cols >= 40 masked). K=64 -> 16 WMMA steps.
// ---------------------------------------------------------------------------
__global__ void __launch_bounds__(96)
lin1_kernel(const float* __restrict__ h, const float* __restrict__ w,
            const float* __restrict__ bias, float* __restrict__ out)
{
    const int tid  = threadIdx.x;
    const int wave = tid >> 5;
    const int lane = tid & 31;
    const int row0 = blockIdx.x * 16;
    const int n0   = wave * 16;

    const int m     = lane & 15;
    const int khalf = lane >> 4;
    const int col   = lane & 15;
    const int n     = n0 + col;
    const bool valid = (n < N_CLASSES);

    v8f acc = {};
#pragma unroll
    for (int k0 = 0; k0 < HIDDEN; k0 += 4) {
        const int ka = k0 + khalf * 2;
        v2f a, b;
        a.x = h[(size_t)(row0 + m) * HIDDEN + ka];
        a.y = h[(size_t)(row0 + m) * HIDDEN + ka + 1];
        b.x = valid ? w[(ka)     * N_CLASSES + n] : 0.0f;
        b.y = valid ? w[(ka + 1) * N_CLASSES + n] : 0.0f;
        acc = __builtin_amdgcn_wmma_f32_16x16x4_f32(false, a, false, b,
                                                    (short)0, acc, false, false);
    }

    const float bv = valid ? bias[n] : 0.0f;
#pragma unroll
    for (int r = 0; r < 8; ++r) {
        const int mrow = r + khalf * 8;
        if (valid)
            out[(size_t)(row0 + mrow) * N_CLASSES + n] = acc[r] + bv;
    }
}

// ---------------------------------------------------------------------------
// Kernel 5: in-place row-wise log_softmax over 40 classes.
// 40 floats/row live in L2 after kernel 4; one thread per row is fine.
// ---------------------------------------------------------------------------
__global__ void logsoftmax_kernel(float* __restrict__ out, int nrows)
{
    const int row = blockIdx.x * blockDim.x + threadIdx.x;
    if (row >= nrows) return;
    float* p = out + (size_t)row * N_CLASSES;
    float mx = -1e30f;
#pragma unroll
    for (int j = 0; j < N_CLASSES; ++j) mx = fmaxf(mx, p[j]);
    float s = 0.0f;
#pragma unroll
    for (int j = 0; j < N_CLASSES; ++j) s += __expf(p[j] - mx);
    const float ls = __logf(s) + mx;
#pragma unroll
    for (int j = 0; j < N_CLASSES; ++j) p[j] = p[j] - ls;
}

// ---------------------------------------------------------------------------
extern "C" void kernel_launch(void* const* d_in, const int* in_sizes, int n_in,
                              void* d_out, int out_size, void* d_ws, size_t ws_size,
                              hipStream_t stream)
{
    const float* x    = (const float*)d_in[0];
    const int*   esrc = (const int*)  d_in[1];
    const int*   edst = (const int*)  d_in[2];
    const float* ew   = (const float*)d_in[3];
    const float* l0w  = (const float*)d_in[4];
    const float* l0b  = (const float*)d_in[5];
    const float* cw   = (const float*)d_in[6];
    const float* l1w  = (const float*)d_in[7];
    const float* l1b  = (const float*)d_in[8];
    float* out = (float*)d_out;

    const int nE = in_sizes[1];
    const size_t nh = (size_t)N_NODES * HIDDEN;

    float* h   = (float*)d_ws;
    float* x0  = h  + nh;
    float* agg = x0 + nh;

    const int rowTiles = N_NODES / 16;   // 6250

    lin0_kernel<<<rowTiles, 128, 0, stream>>>(x, l0w, l0b, h, x0);

    for (int l = 0; l < N_LAYERS; ++l) {
        const float beta = logf(0.5f / (float)(l + 1) + 1.0f);
        zero_kernel<<<1024, 256, 0, stream>>>(agg, (int)nh);
        spmm_kernel<<<1024, 256, 0, stream>>>(esrc, edst, ew, h, agg, nE);
        layer_kernel<<<rowTiles, 128, 0, stream>>>(agg, x0,
                                                   cw + (size_t)l * HIDDEN * HIDDEN,
                                                   h, beta);
    }

    lin1_kernel<<<rowTiles, 96, 0, stream>>>(h, l1w, l1b, out);
    logsoftmax_kernel<<<(N_NODES + 255) / 256, 256, 0, stream>>>(out, N_NODES);
}